// NaiveDynMessage_77979426226457
// MI455X (gfx1250) — compile-verified
//
#include <hip/hip_runtime.h>

// CDNA5 / gfx1250. wave32. f32 WMMA (V_WMMA_F32_16X16X4_F32).
typedef __attribute__((ext_vector_type(2))) float v2f;
typedef __attribute__((ext_vector_type(8))) float v8f;

#define DHV 128          // hv feature dim == MV == ME
#define PITCH 132        // LDS row pitch (floats): 132%64==4 -> conflict-free A-frag reads

// ---------------- small prep kernels ----------------

__global__ void k_zero_flags(int* __restrict__ flag, int n) {
  int i = blockIdx.x * blockDim.x + threadIdx.x;
  if (i < n) flag[i] = 0;
}

// every endpoint of any edge is a softmax destination (edges are bidirectionalized)
__global__ void k_mark(const int* __restrict__ eu, const int* __restrict__ ev,
                       int* __restrict__ flag, int E) {
  int e = blockIdx.x * blockDim.x + threadIdx.x;
  if (e < E) { flag[eu[e]] = 1; flag[ev[e]] = 1; }
}

// Stage both weight matrices TRANSPOSED (column-major) so a B fragment
// (B[k][col], B[k+1][col]) is one contiguous 8-byte load in the hot loops.
//   WaT[j*128+i]  = Wa[i*128+j]
//   WlT[j*128+i]  = Wl[i*128+j] + Wl[(i+134)*128+j]   (p/q blocks cancel in folded GEMM)
__global__ void k_prep(const float* __restrict__ Wa, const float* __restrict__ Wl,
                       float* __restrict__ WaT, float* __restrict__ WlT) {
  int t = blockIdx.x * blockDim.x + threadIdx.x;   // 16384 exactly
  int i = t >> 7, j = t & 127;
  WaT[j * 128 + i] = Wa[t];
  WlT[j * 128 + i] = Wl[t] + Wl[t + 134 * 128];
}

// ---------------- mv kernel: mv[n] = flag ? elu(leaky_relu(hv[n] @ Wa + ba)) : 0 ----------------
// block = 256 thr = 8 waves; block tile = 16 vertices x 128 cols; wave owns 16 cols.
__global__ __launch_bounds__(256) void k_mv(const float* __restrict__ hv,
                                            const float* __restrict__ WaT,
                                            const float* __restrict__ ba,
                                            const int* __restrict__ flag,
                                            float* __restrict__ out, int N) {
  __shared__ float sA[16 * PITCH];
  __shared__ int   sFlag[16];
  const int tid  = threadIdx.x;
  const int lane = tid & 31;
  const int wave = tid >> 5;
  const int n0   = blockIdx.x * 16;

  // cooperative load of 16 contiguous hv rows into LDS (padded pitch) + flags
  const float4* hv4 = reinterpret_cast<const float4*>(hv);
  for (int f = tid; f < 16 * 32; f += 256) {
    int r = f >> 5, c4 = f & 31;
    int n = n0 + r; if (n >= N) n = N - 1;
    float4 v = hv4[(size_t)n * 32 + c4];
    float* dst = &sA[r * PITCH + c4 * 4];
    dst[0] = v.x; dst[1] = v.y; dst[2] = v.z; dst[3] = v.w;
  }
  if (tid < 16) {
    int n = n0 + tid;
    sFlag[tid] = (n < N) ? flag[n] : 0;
  }
  __syncthreads();

  const int hi  = lane >> 4;     // 0: K=k,k+1   1: K=k+2,k+3
  const int ln  = lane & 15;
  const int col = wave * 16 + ln;

  v8f acc = {};
#pragma unroll 8
  for (int k = 0; k < DHV; k += 4) {
    v2f a = *reinterpret_cast<const v2f*>(&sA[ln * PITCH + k + 2 * hi]);
    v2f b = *reinterpret_cast<const v2f*>(&WaT[col * 128 + k + 2 * hi]);
    acc = __builtin_amdgcn_wmma_f32_16x16x4_f32(false, a, false, b, (short)0, acc,
                                                false, false);
  }

  const float bias = ba[col];
  if (n0 + 16 <= N) {                               // uniform fast path (always for N=50000)
    float* p = out + (size_t)(n0 + 8 * hi) * DHV + col;
#pragma unroll
    for (int g = 0; g < 8; ++g) {
      float x = acc[g] + bias;
      x = x > 0.f ? x : 0.01f * x;                  // leaky_relu
      x = x > 0.f ? x : (__expf(x) - 1.f);          // elu
      x = sFlag[g + 8 * hi] ? x : 0.f;              // isolated vertex: empty segment
      __builtin_nontemporal_store(x, p + (size_t)g * DHV);
    }
  } else {
#pragma unroll
    for (int g = 0; g < 8; ++g) {
      int n = n0 + g + 8 * hi;
      if (n < N) {
        float x = acc[g] + bias;
        x = x > 0.f ? x : 0.01f * x;
        x = x > 0.f ? x : (__expf(x) - 1.f);
        x = sFlag[g + 8 * hi] ? x : 0.f;
        out[(size_t)n * DHV + col] = x;
      }
    }
  }
}

// ---------------- me kernel: me[e] = leaky_relu((hv[u]+hv[v]) @ W' + 2*bl) ----------------
// block = 256 thr = 8 waves; block tile = 64 edges x 128 cols (4 row tiles per wave).
__global__ __launch_bounds__(256) void k_me(const float* __restrict__ hv,
                                            const int* __restrict__ eu,
                                            const int* __restrict__ ev,
                                            const float* __restrict__ WlT,
                                            const float* __restrict__ bl,
                                            float* __restrict__ me, int E) {
  __shared__ float sS[64 * PITCH];   // 33.8 KB: s = hv[u]+hv[v] rows
  const int tid  = threadIdx.x;
  const int lane = tid & 31;
  const int wave = tid >> 5;
  const int e0   = blockIdx.x * 64;

  // gather phase: each wave builds 8 rows; one row = 32 lanes x float4
  const float4* hv4 = reinterpret_cast<const float4*>(hv);
#pragma unroll
  for (int i = 0; i < 8; ++i) {
    int el = wave * 8 + i;
    int e = e0 + el; if (e >= E) e = E - 1;
    int u = eu[e], v = ev[e];                       // scalar (broadcast) loads
    float4 a = hv4[(size_t)u * 32 + lane];
    float4 b = hv4[(size_t)v * 32 + lane];
    float* dst = &sS[el * PITCH + lane * 4];
    dst[0] = a.x + b.x; dst[1] = a.y + b.y; dst[2] = a.z + b.z; dst[3] = a.w + b.w;
  }
  __syncthreads();

  const int hi  = lane >> 4;
  const int ln  = lane & 15;
  const int col = wave * 16 + ln;

  v8f acc[4] = {v8f{}, v8f{}, v8f{}, v8f{}};
#pragma unroll 4
  for (int k = 0; k < DHV; k += 4) {
    v2f b = *reinterpret_cast<const v2f*>(&WlT[col * 128 + k + 2 * hi]);  // shared by 4 tiles
#pragma unroll
    for (int t = 0; t < 4; ++t) {
      v2f a = *reinterpret_cast<const v2f*>(&sS[(t * 16 + ln) * PITCH + k + 2 * hi]);
      acc[t] = __builtin_amdgcn_wmma_f32_16x16x4_f32(false, a, false, b, (short)0,
                                                     acc[t], false, false);
    }
  }

  const float bias = 2.f * bl[col];                 // bl added in both directions
  if (e0 + 64 <= E) {                               // uniform fast path (always for E=400000)
#pragma unroll
    for (int t = 0; t < 4; ++t) {
      float* p = me + (size_t)(e0 + t * 16 + 8 * hi) * DHV + col;
#pragma unroll
      for (int g = 0; g < 8; ++g) {
        float x = acc[t][g] + bias;
        x = x > 0.f ? x : 0.01f * x;                // leaky_relu
        __builtin_nontemporal_store(x, p + (size_t)g * DHV);  // stream out, keep L2 for hv
      }
    }
  } else {
#pragma unroll
    for (int t = 0; t < 4; ++t) {
#pragma unroll
      for (int g = 0; g < 8; ++g) {
        int e = e0 + t * 16 + g + 8 * hi;
        if (e < E) {
          float x = acc[t][g] + bias;
          me[(size_t)e * DHV + col] = x > 0.f ? x : 0.01f * x;
        }
      }
    }
  }
}

// ---------------- host ----------------

extern "C" void kernel_launch(void* const* d_in, const int* in_sizes, int n_in,
                              void* d_out, int out_size, void* d_ws, size_t ws_size,
                              hipStream_t stream) {
  const float* hv = (const float*)d_in[0];
  // d_in[1] he_ftr, d_in[2] p_ftr, d_in[3] q_ftr, d_in[8] Wal, d_in[9] bal:
  // provably unused in the forward value (softmax weights sum to 1; p/q cancel).
  const int*   eu = (const int*)d_in[4];
  const int*   ev = (const int*)d_in[5];
  const float* Wa = (const float*)d_in[6];
  const float* ba = (const float*)d_in[7];
  const float* Wl = (const float*)d_in[10];
  const float* bl = (const float*)d_in[11];

  const int N = in_sizes[0] / DHV;   // 50000
  const int E = in_sizes[4];         // 400000

  int*   flag = (int*)d_ws;
  size_t flag_bytes = (((size_t)N * sizeof(int)) + 255) & ~(size_t)255;
  float* WaT = (float*)((char*)d_ws + flag_bytes);          // 64 KB
  float* WlT = WaT + 128 * 128;                             // 64 KB

  float* mv = (float*)d_out;
  float* me = (float*)d_out + (size_t)N * DHV;

  k_zero_flags<<<(N + 255) / 256, 256, 0, stream>>>(flag, N);
  k_mark<<<(E + 255) / 256, 256, 0, stream>>>(eu, ev, flag, E);
  k_prep<<<64, 256, 0, stream>>>(Wa, Wl, WaT, WlT);
  k_mv<<<(N + 15) / 16, 256, 0, stream>>>(hv, WaT, ba, flag, mv, N);
  k_me<<<(E + 63) / 64, 256, 0, stream>>>(hv, eu, ev, WlT, bl, me, E);
}